// MultiEmbodimentActionEncoder_19954418057652
// MI455X (gfx1250) — compile-verified
//
#include <hip/hip_runtime.h>
#include <math.h>

// ---------------------------------------------------------------------------
// MultiEmbodimentActionEncoder on gfx1250 (MI455X).
// fp32 GEMMs via V_WMMA_F32_16X16X4_F32; global->LDS staging via
// GLOBAL_LOAD_ASYNC_TO_LDS_B128 (ASYNCcnt) with double-buffered LDS tiles.
// B=256 batches x H=64 tokens, D=64, hid=emb=1024, C=8 categories.
// ---------------------------------------------------------------------------

typedef float v2f __attribute__((ext_vector_type(2)));
typedef float v8f __attribute__((ext_vector_type(8)));

#define TILE_M 64
#define TILE_N 128
#define TILE_K 64
#define LDA 68    // 64 + 4: A-frag b64 loads hit all 64 banks once; rows 16B-aligned
#define LDB 136   // 128 + 8: half-wave B-frag rows hit disjoint bank groups; 16B-aligned
#define THREADS 256

// ---- async global -> LDS staging (CDNA5 TENSOR/ASYNC path) ----------------
// GVS form: mem = SADDR(64) + VADDR(32) + IOFFSET ; LDS dest from vdst VGPR.
__device__ __forceinline__ void stage_tiles_async(const float* __restrict__ Xr,
                                                  const float* __restrict__ Wc,
                                                  int K, int NOUT, int k0, int tid,
                                                  unsigned ldsA, unsigned ldsB) {
  const unsigned long long xb = (unsigned long long)Xr;
  const unsigned long long wb = (unsigned long long)Wc;
#pragma unroll
  for (int i = 0; i < 4; ++i) {            // A: 64x64 floats = 1024 b128 chunks
    int idx = tid + i * THREADS;
    int r = idx >> 4, cc = (idx & 15) * 4;
    unsigned lds  = ldsA + (unsigned)(r * LDA + cc) * 4u;
    unsigned goff = (unsigned)((r * K + k0 + cc) * 4);
    asm volatile("global_load_async_to_lds_b128 %0, %1, %2"
                 :: "v"(lds), "v"(goff), "s"(xb) : "memory");
  }
#pragma unroll
  for (int i = 0; i < 8; ++i) {            // B: 64x128 floats = 2048 b128 chunks
    int idx = tid + i * THREADS;
    int r = idx >> 5, cc = (idx & 31) * 4;
    unsigned lds  = ldsB + (unsigned)(r * LDB + cc) * 4u;
    unsigned goff = (unsigned)(((k0 + r) * NOUT + cc) * 4);
    asm volatile("global_load_async_to_lds_b128 %0, %1, %2"
                 :: "v"(lds), "v"(goff), "s"(wb) : "memory");
  }
}

__device__ __forceinline__ void wait_async_all() {
  asm volatile("s_wait_asynccnt 0x0" ::: "memory");
}

// ---- 16x64 wave tile, K=64 block, f32 WMMA --------------------------------
__device__ __forceinline__ void mma_block(const float* __restrict__ a,
                                          const float* __restrict__ b,
                                          int wm, int wn, int l16, int half,
                                          v8f (&acc)[4]) {
#pragma unroll
  for (int k = 0; k < TILE_K; k += 4) {
    // A frag 16x4: lanes 0-15 rows m with K=k..k+1; lanes 16-31 K=k+2..k+3
    v2f av = *(const v2f*)(a + (wm + l16) * LDA + k + 2 * half);
    const float* bp = b + (k + 2 * half) * LDB + wn + l16;
#pragma unroll
    for (int n = 0; n < 4; ++n) {
      // B frag 4x16: v0 = B[k(+2h)][col], v1 = B[k(+2h)+1][col]
      v2f bv;
      bv.x = bp[n * 16];
      bv.y = bp[n * 16 + LDB];
      acc[n] = __builtin_amdgcn_wmma_f32_16x16x4_f32(
          /*neg_a=*/false, av, /*neg_b=*/false, bv,
          /*c_mod=*/(short)0, acc[n], /*reuse_a=*/false, /*reuse_b=*/false);
    }
  }
}

template <int RELU, int PE>
__global__ __launch_bounds__(THREADS) void
cat_linear_wmma(const float* __restrict__ X,   // [Nrows, K]
                const float* __restrict__ W,   // [C, K, NOUT]
                const float* __restrict__ Bias,// [C, NOUT]
                const int*  __restrict__ Cat,  // [B]
                float* __restrict__ Y,         // [Nrows, NOUT]
                int K, int NOUT) {
  __shared__ float sA[2][TILE_M * LDA];
  __shared__ float sB[2][TILE_K * LDB];

  const int tid  = threadIdx.x;
  const int lane = tid & 31;
  const int wave = tid >> 5;
  const int l16  = lane & 15;
  const int half = lane >> 4;
  const int wm   = (wave & 3) * 16;   // wave's M offset inside the 64-row tile
  const int wn   = (wave >> 2) * 64;  // wave's N offset inside the 128-col tile

  const int batch   = blockIdx.x;
  const int rowBase = batch * TILE_M;        // 64 tokens of one batch
  const int colBase = blockIdx.y * TILE_N;
  const int c       = Cat[batch];            // one category per batch

  const float* Xr = X + (size_t)rowBase * K;
  const float* Wc = W + (size_t)c * K * NOUT + colBase;

  // Generic __shared__ addresses are {SHARED_BASE, lds_offset}: low 32 bits are
  // the workgroup-relative LDS byte offset the async op needs in its vdst VGPR.
  const unsigned ldsA[2] = {(unsigned)(uintptr_t)&sA[0][0],
                            (unsigned)(uintptr_t)&sA[1][0]};
  const unsigned ldsB[2] = {(unsigned)(uintptr_t)&sB[0][0],
                            (unsigned)(uintptr_t)&sB[1][0]};

  v8f acc[4] = {v8f{}, v8f{}, v8f{}, v8f{}};

  stage_tiles_async(Xr, Wc, K, NOUT, 0, tid, ldsA[0], ldsB[0]);
  wait_async_all();
  __syncthreads();

  const int ktiles = K / TILE_K;
  for (int kt = 0; kt < ktiles; ++kt) {
    const int cur = kt & 1;
    if (kt + 2 < ktiles)  // speculative L2 prefetch of the tile after next
      __builtin_prefetch(Wc + (size_t)(kt + 2) * TILE_K * NOUT, 0, 0);
    if (kt + 1 < ktiles)  // DMA next tile into the other buffer behind the math
      stage_tiles_async(Xr, Wc, K, NOUT, (kt + 1) * TILE_K, tid,
                        ldsA[cur ^ 1], ldsB[cur ^ 1]);
    mma_block(sA[cur], sB[cur], wm, wn, l16, half, acc);
    if (kt + 1 < ktiles) wait_async_all();  // own wave's async writes landed
    __syncthreads();                        // cross-wave visibility
  }

  // ---- epilogue: bias (+relu) (+sinusoidal PE, layer 1 only) --------------
  const float NEG_LOG1E4_OVER_DIM = -9.210340371976184f / 1024.0f; // -ln(1e4)/hid
#pragma unroll
  for (int n = 0; n < 4; ++n) {
    const int col = colBase + wn + n * 16 + l16;
    float bia = Bias[(size_t)c * NOUT + col];
#pragma unroll
    for (int j = 0; j < 8; ++j) {
      const int m   = wm + j + half * 8;     // row inside the 64-token tile
      const int row = rowBase + m;
      float v = acc[n][j] + bia;
      if (RELU) v = fmaxf(v, 0.0f);
      if (PE) {
        // pe[t, 2i] = sin(t*f_i), pe[t, 2i+1] = cos(t*f_i),
        // f_i = exp(-ln(1e4)*2i/hid); t = position within H (== m)
        float freq = __expf((float)(col & ~1) * NEG_LOG1E4_OVER_DIM);
        float ang  = (float)m * freq;
        v += (col & 1) ? __cosf(ang) : __sinf(ang);
      }
      Y[(size_t)row * NOUT + col] = v;
    }
  }
}

// ---------------------------------------------------------------------------
extern "C" void kernel_launch(void* const* d_in, const int* in_sizes, int n_in,
                              void* d_out, int out_size, void* d_ws, size_t ws_size,
                              hipStream_t stream) {
  (void)in_sizes; (void)n_in; (void)out_size; (void)ws_size;

  const float* x   = (const float*)d_in[0];  // [256, 64, 64]
  const int*   cat = (const int*)  d_in[1];  // [256]
  const float* W1  = (const float*)d_in[2];  // [8, 64, 1024]
  const float* b1  = (const float*)d_in[3];  // [8, 1024]
  const float* W2  = (const float*)d_in[4];  // [8, 1024, 1024]
  const float* b2  = (const float*)d_in[5];  // [8, 1024]
  const float* W3  = (const float*)d_in[6];  // [8, 1024, 1024]
  const float* b3  = (const float*)d_in[7];  // [8, 1024]
  float* out = (float*)d_out;                // [256, 64, 1024]

  float* y1 = (float*)d_ws;                       // [16384, 1024]
  float* y2 = y1 + (size_t)256 * 64 * 1024;       // [16384, 1024]

  dim3 block(THREADS);
  dim3 grid(256, 1024 / TILE_N);   // one batch (64 rows, 1 category) x 8 col tiles

  hipLaunchKernelGGL((cat_linear_wmma<1, 1>), grid, block, 0, stream,
                     x,  W1, b1, cat, y1, 64,   1024);
  hipLaunchKernelGGL((cat_linear_wmma<1, 0>), grid, block, 0, stream,
                     y1, W2, b2, cat, y2, 1024, 1024);
  hipLaunchKernelGGL((cat_linear_wmma<0, 0>), grid, block, 0, stream,
                     y2, W3, b3, cat, out, 1024, 1024);
}